// FFM_69664369541798
// MI455X (gfx1250) — compile-verified
//
#include <hip/hip_runtime.h>
#include <stdint.h>

typedef float f4  __attribute__((ext_vector_type(4)));
typedef float v2f __attribute__((ext_vector_type(2)));
typedef float v8f __attribute__((ext_vector_type(8)));

#define BLOCK 128

// LDS layout (float offsets). All table bases are multiples of 4 floats (16B).
#define OFF_AU   0      // 64
#define OFF_AI   64     // 64
#define OFF_GU   128    // 128
#define OFF_GI   256    // 128
#define OFF_OU   384    // 1344
#define OFF_OI   1728   // 1344
#define OFF_MU   3072   // 1216
#define OFF_MI   4288   // 1216
#define OFF_LIN  5504   // 44 (43 weights + 1 zero pad)
#define OFF_F    5552   // 128*45 = 5760 feature tile
#define OFF_LOUT 11312  // 128 per-sample dense-lin results from WMMA
#define SMEM_FLOATS 11440

__device__ __forceinline__ void async_cp16(unsigned lds_byte_off, const void* gptr) {
  // GLOBAL_LOAD_ASYNC_TO_LDS_B128: vdst = LDS byte address, vaddr = 64-bit global address
  asm volatile("global_load_async_to_lds_b128 %0, %1, off"
               :: "v"(lds_byte_off), "v"((unsigned long long)(uintptr_t)gptr)
               : "memory");
}

__global__ __launch_bounds__(BLOCK) void ffm_kernel(
    const float* __restrict__ fv,
    const float* __restrict__ age_u,  const float* __restrict__ age_i,
    const float* __restrict__ gen_u,  const float* __restrict__ gen_i,
    const float* __restrict__ occ_u,  const float* __restrict__ occ_i,
    const float* __restrict__ mov_u,  const float* __restrict__ mov_i,
    const float* __restrict__ uid_u,  const float* __restrict__ uid_i,
    const float* __restrict__ iid_u,  const float* __restrict__ iid_i,
    const float* __restrict__ user_w, const float* __restrict__ item_w,
    const float* __restrict__ lin_w,  const float* __restrict__ lin_b,
    float* __restrict__ out)
{
  __shared__ float smem[SMEM_FLOATS];
  const int tid = threadIdx.x;
  const unsigned sbase = (unsigned)(uintptr_t)(&smem[0]);

  // ---- Stage small tables + this block's feature tile into LDS (async, no VGPR round trip)
  auto stage = [&](int dstF, const float* src, int n4) {
    for (int i = tid; i < n4; i += BLOCK)
      async_cp16(sbase + (unsigned)(dstF + i * 4) * 4u, src + i * 4);
  };
  stage(OFF_AU, age_u, 16);
  stage(OFF_AI, age_i, 16);
  stage(OFF_GU, gen_u, 32);
  stage(OFF_GI, gen_i, 32);
  stage(OFF_OU, occ_u, 336);
  stage(OFF_OI, occ_i, 336);
  stage(OFF_MU, mov_u, 304);
  stage(OFF_MI, mov_i, 304);
  stage(OFF_F,  fv + (size_t)blockIdx.x * BLOCK * 45, (BLOCK * 45) / 4);
  if (tid < 43) smem[OFF_LIN + tid] = lin_w[tid];
  if (tid == 43) smem[OFF_LIN + 43] = 0.0f;  // K pad
  asm volatile("s_wait_asynccnt 0" ::: "memory");
  __syncthreads();

  // ---- WMMA: dense linear term  lind[s] = sum_k F[s][2+k] * lin_w[k]  (k<43), exact f32.
  // Broadcast-B trick: B[k][n] = w[k] for all n  =>  every column of D holds the matvec.
  {
    const int wave = tid >> 5;
    const int lane = tid & 31;
    const int koff = (lane >= 16) ? 2 : 0;
    for (int h = 0; h < 2; ++h) {
      const int r0  = wave * 32 + h * 16;
      const int row = r0 + (lane & 15);
      v8f accw = {0.f, 0.f, 0.f, 0.f, 0.f, 0.f, 0.f, 0.f};
#pragma unroll
      for (int k0 = 0; k0 < 44; k0 += 4) {
        const int kb = k0 + koff;           // lanes 0-15: k0, lanes 16-31: k0+2
        v2f a, b;
        a.x = smem[OFF_F + row * 45 + 2 + kb];                       // kb <= 42 always valid
        a.y = (kb + 1 < 43) ? smem[OFF_F + row * 45 + 3 + kb] : 0.f; // pad k=43
        b.x = smem[OFF_LIN + kb];
        b.y = smem[OFF_LIN + kb + 1];                                 // LIN[43]=0
        accw = __builtin_amdgcn_wmma_f32_16x16x4_f32(
            false, a, false, b, (short)0, accw, false, false);
      }
      if ((lane & 15) == 0) {               // lanes 0 / 16 hold column N=0: rows r0..r0+7 / r0+8..r0+15
        const int mbase = r0 + ((lane >= 16) ? 8 : 0);
#pragma unroll
        for (int r = 0; r < 8; ++r) smem[OFF_LOUT + mbase + r] = accw[r];
      }
    }
  }

  // ---- Per-thread: parse feature row
  const float* Frow = &smem[OFF_F + tid * 45];
  const int   uid = (int)Frow[0];
  const int   iid = (int)Frow[1];
  const float age = Frow[2];
  const int   g   = (Frow[3] > 0.5f) ? 0 : 1;
  int o = 0;
#pragma unroll
  for (int j = 1; j < 21; ++j) o = (Frow[5 + j] > 0.5f) ? j : o;
  unsigned mbits = 0;
#pragma unroll
  for (int j = 0; j < 19; ++j) mbits |= (Frow[26 + j] > 0.5f) ? (1u << j) : 0u;

  // ---- cross = gu.au + ou.(au+gu) + uu.(au+gu+ou) + (mu+tu).(ai+gi+oi+ui) + mi.ti
  const f4* AU = (const f4*)&smem[OFF_AU];
  const f4* AI = (const f4*)&smem[OFF_AI];
  const f4* GU = (const f4*)&smem[OFF_GU + g * 64];
  const f4* GI = (const f4*)&smem[OFF_GI + g * 64];
  const f4* OU = (const f4*)&smem[OFF_OU + o * 64];
  const f4* OI = (const f4*)&smem[OFF_OI + o * 64];
  const f4* MU = (const f4*)&smem[OFF_MU];
  const f4* MI = (const f4*)&smem[OFF_MI];
  const f4* UU = (const f4*)(uid_u + (size_t)uid * 64);
  const f4* UI = (const f4*)(uid_i + (size_t)uid * 64);
  const f4* TU = (const f4*)(iid_u + (size_t)iid * 64);
  const f4* TI = (const f4*)(iid_i + (size_t)iid * 64);

  f4 acc = {0.f, 0.f, 0.f, 0.f};
#pragma unroll 4
  for (int v = 0; v < 16; ++v) {
    f4 au = age * AU[v];
    f4 ai = age * AI[v];
    f4 gu = GU[v], gi = GI[v], ou = OU[v], oi = OI[v];
    f4 mu = {0.f, 0.f, 0.f, 0.f}, mi = {0.f, 0.f, 0.f, 0.f};
    unsigned mb = mbits;
    while (mb) {                      // ~3.8 set bits on average
      const int j = __builtin_ctz(mb);
      mb &= mb - 1;
      mu += MU[j * 16 + v];
      mi += MI[j * 16 + v];
    }
    f4 uu = UU[v], ui = UI[v], tu = TU[v], ti = TI[v];   // L2-resident gathers
    f4 p2 = au + gu;
    f4 p3 = p2 + ou;
    f4 r  = ai + gi + oi + ui;
    acc += gu * au + ou * p2 + uu * p3 + (mu + tu) * r + mi * ti;
  }
  const float cross = acc.x + acc.y + acc.z + acc.w;

  float wsum = 0.f;
#pragma unroll
  for (int j = 0; j < 43; ++j) wsum += smem[OFF_LIN + j];

  __syncthreads();                    // WMMA lin results visible
  const float lind = smem[OFF_LOUT + tid];

  const float z = user_w[uid] + item_w[iid] + lind + cross * wsum + lin_b[0];
  out[(size_t)blockIdx.x * BLOCK + tid] = 1.0f / (1.0f + __expf(-z));
}

extern "C" void kernel_launch(void* const* d_in, const int* in_sizes, int n_in,
                              void* d_out, int out_size, void* d_ws, size_t ws_size,
                              hipStream_t stream) {
  const float* fv     = (const float*)d_in[0];
  const float* age_u  = (const float*)d_in[1];
  const float* age_i  = (const float*)d_in[2];
  const float* gen_u  = (const float*)d_in[3];
  const float* gen_i  = (const float*)d_in[4];
  const float* occ_u  = (const float*)d_in[5];
  const float* occ_i  = (const float*)d_in[6];
  const float* mov_u  = (const float*)d_in[7];
  const float* mov_i  = (const float*)d_in[8];
  const float* uid_u  = (const float*)d_in[9];
  const float* uid_i  = (const float*)d_in[10];
  const float* iid_u  = (const float*)d_in[11];
  const float* iid_i  = (const float*)d_in[12];
  const float* user_w = (const float*)d_in[13];
  const float* item_w = (const float*)d_in[14];
  const float* lin_w  = (const float*)d_in[15];
  const float* lin_b  = (const float*)d_in[16];
  float* out = (float*)d_out;

  const int B = in_sizes[0] / 45;          // 262144
  dim3 grid(B / BLOCK), block(BLOCK);
  ffm_kernel<<<grid, block, 0, stream>>>(fv, age_u, age_i, gen_u, gen_i,
                                         occ_u, occ_i, mov_u, mov_i,
                                         uid_u, uid_i, iid_u, iid_i,
                                         user_w, item_w, lin_w, lin_b, out);
}